// DMGI_89429809037887
// MI455X (gfx1250) — compile-verified
//
#include <hip/hip_runtime.h>

// ---------------------------------------------------------------------------
// DMGI forward on MI455X (gfx1250, wave32).
// pos_h[r] = relu(GCNConv(x, edges_r));  neg_h[r] = relu(GCNConv(x[perm_r], edges_r))
// Uses  x[perm] @ W == (x @ W)[perm]  so only one WMMA GEMM per relation.
// Dense GEMM: V_WMMA_F32_16X16X4_F32 (full f32 precision, 16x64 tile per wave).
// Scatter (bandwidth/atomic bound, dominates runtime) uses f32 global atomics.
// ---------------------------------------------------------------------------

#define N_NODES 100000
#define N_EDGES 600000
#define N_REL   3
#define F_IN    128
#define F_OUT   64
#define M_TILES (N_NODES / 16)   // 6250, exact

typedef float v2f __attribute__((ext_vector_type(2)));
typedef float v8f __attribute__((ext_vector_type(8)));

// --- degree / normalization -------------------------------------------------

__global__ void dmgi_deg_init(float* __restrict__ dinv) {
    int i = blockIdx.x * blockDim.x + threadIdx.x;
    if (i < N_NODES) dinv[i] = 1.0f;            // self-loop contributes 1
}

__global__ void dmgi_deg_count(const int* __restrict__ dst, float* __restrict__ dinv) {
    int e = blockIdx.x * blockDim.x + threadIdx.x;
    if (e < N_EDGES) atomicAdd(&dinv[dst[e]], 1.0f);
}

__global__ void dmgi_deg_rsqrt(float* __restrict__ dinv) {
    int i = blockIdx.x * blockDim.x + threadIdx.x;
    if (i < N_NODES) {
        float d = dinv[i];
        dinv[i] = (d > 0.0f) ? rsqrtf(d) : 0.0f;
    }
}

// --- dense GEMM  hw = x @ W  via V_WMMA_F32_16X16X4_F32 ---------------------
// One wave -> 16 rows x 64 cols. K looped in steps of 4.
// A frag (16x4 f32): lanes 0-15 hold K={0,1}, lanes 16-31 hold K={2,3}; M=lane&15.
// B frag (4x16 f32): same K split; N=lane&15.
// C/D (16x16 f32, 8 VGPRs): vgpr v = rows v (lanes 0-15) and v+8 (lanes 16-31).

__global__ __launch_bounds__(256) void dmgi_gemm_wmma(const float* __restrict__ x,
                                                      const float* __restrict__ W,
                                                      float* __restrict__ hw) {
    int wave = (int)((blockIdx.x * 256u + threadIdx.x) >> 5);
    if (wave >= M_TILES) return;                 // wave-uniform: EXEC stays all-ones
    int lane = threadIdx.x & 31;
    int half = lane >> 4;                        // 0 -> K{0,1}, 1 -> K{2,3}
    int lrow = lane & 15;                        // M row for A, N col for B

    const float* xp  = x + (size_t)(wave * 16 + lrow) * F_IN + 2 * half;
    const float* wpb = W + (size_t)(2 * half) * F_OUT + lrow;

    v8f acc0 = {}, acc1 = {}, acc2 = {}, acc3 = {};

    #pragma unroll 4
    for (int k0 = 0; k0 < F_IN; k0 += 4) {
        v2f a = *(const v2f*)(xp + k0);
        const float* wp = wpb + (size_t)k0 * F_OUT;
        v2f b0 = { wp[ 0], wp[ 0 + F_OUT] };
        v2f b1 = { wp[16], wp[16 + F_OUT] };
        v2f b2 = { wp[32], wp[32 + F_OUT] };
        v2f b3 = { wp[48], wp[48 + F_OUT] };
        acc0 = __builtin_amdgcn_wmma_f32_16x16x4_f32(false, a, false, b0, (short)0, acc0, false, false);
        acc1 = __builtin_amdgcn_wmma_f32_16x16x4_f32(false, a, false, b1, (short)0, acc1, false, false);
        acc2 = __builtin_amdgcn_wmma_f32_16x16x4_f32(false, a, false, b2, (short)0, acc2, false, false);
        acc3 = __builtin_amdgcn_wmma_f32_16x16x4_f32(false, a, false, b3, (short)0, acc3, false, false);
    }

    // store: row m = wave*16 + half*8 + v, col = ntile*16 + lrow
    float* op = hw + (size_t)(wave * 16 + half * 8) * F_OUT + lrow;
    #pragma unroll
    for (int v = 0; v < 8; ++v) {
        op[(size_t)v * F_OUT +  0] = acc0[v];
        op[(size_t)v * F_OUT + 16] = acc1[v];
        op[(size_t)v * F_OUT + 32] = acc2[v];
        op[(size_t)v * F_OUT + 48] = acc3[v];
    }
}

// --- init: bias + self-loop contribution (norm = dinv[i]^2), zero summary ---

__global__ void dmgi_init_out(const float* __restrict__ hw, const float* __restrict__ dinv,
                              const int* __restrict__ perm, const float* __restrict__ b,
                              float* __restrict__ pos, float* __restrict__ neg,
                              float* __restrict__ summ) {
    int idx = blockIdx.x * blockDim.x + threadIdx.x;   // over N*F_OUT
    if (idx >= N_NODES * F_OUT) return;
    int i = idx >> 6;
    int f = idx & 63;
    float di = dinv[i];
    float nrm = di * di;
    float bf = b[f];
    pos[idx] = bf + hw[idx] * nrm;
    neg[idx] = bf + hw[(size_t)perm[i] * F_OUT + f] * nrm;
    if (idx < F_OUT) summ[idx] = 0.0f;
}

// --- edge scatter: pos and neg in one pass ----------------------------------

__global__ void dmgi_scatter(const float* __restrict__ hw, const float* __restrict__ dinv,
                             const int* __restrict__ src, const int* __restrict__ dst,
                             const int* __restrict__ perm,
                             float* __restrict__ pos, float* __restrict__ neg) {
    long long idx = (long long)blockIdx.x * blockDim.x + threadIdx.x;  // E * F_OUT
    if (idx >= (long long)N_EDGES * F_OUT) return;
    int e = (int)(idx >> 6);
    int f = (int)(idx & 63);
    int s = src[e];
    int d = dst[e];
    float nrm = dinv[s] * dinv[d];
    atomicAdd(&pos[(size_t)d * F_OUT + f], hw[(size_t)s * F_OUT + f] * nrm);
    atomicAdd(&neg[(size_t)d * F_OUT + f], hw[(size_t)perm[s] * F_OUT + f] * nrm);
}

// --- relu (neg branch) ------------------------------------------------------

__global__ void dmgi_relu(float* __restrict__ p, int n) {
    int i = blockIdx.x * blockDim.x + threadIdx.x;
    if (i < n) p[i] = fmaxf(p[i], 0.0f);
}

// --- relu (pos branch) fused with column mean -> summary --------------------
// 64 blocks, 256 threads: thread (j = tid/64, f = tid%64) walks rows
// i = blockIdx*4 + j + iter*256 (contiguous 1KB per block per iter).

__global__ __launch_bounds__(256) void dmgi_relu_pos_summary(float* __restrict__ pos,
                                                             float* __restrict__ summ) {
    __shared__ float red[4][F_OUT];
    int f = threadIdx.x & 63;
    int j = threadIdx.x >> 6;
    float sum = 0.0f;
    for (int i = blockIdx.x * 4 + j; i < N_NODES; i += 64 * 4) {
        size_t idx = (size_t)i * F_OUT + f;
        float v = fmaxf(pos[idx], 0.0f);
        pos[idx] = v;
        sum += v;
    }
    red[j][f] = sum;
    __syncthreads();
    if (j == 0) {
        float t = red[0][f] + red[1][f] + red[2][f] + red[3][f];
        atomicAdd(&summ[f], t * (1.0f / (float)N_NODES));
    }
}

// ---------------------------------------------------------------------------

extern "C" void kernel_launch(void* const* d_in, const int* in_sizes, int n_in,
                              void* d_out, int out_size, void* d_ws, size_t ws_size,
                              hipStream_t stream) {
    (void)in_sizes; (void)n_in; (void)out_size; (void)ws_size;

    const float* x     = (const float*)d_in[0];                 // [N, F_IN]
    const int*   edges = (const int*)d_in[1];                   // [R, 2, E]
    const int*   perms = (const int*)d_in[2];                   // [R, N]
    const float* Ws    = (const float*)d_in[3];                 // [R, F_IN, F_OUT]
    const float* bs    = (const float*)d_in[4];                 // [R, F_OUT]

    float* out      = (float*)d_out;
    float* pos_base = out;                                              // [R, N, F_OUT]
    float* neg_base = out + (size_t)N_REL * N_NODES * F_OUT;            // [R, N, F_OUT]
    float* sum_base = out + (size_t)2 * N_REL * N_NODES * F_OUT;        // [R, 1, F_OUT]

    float* hw   = (float*)d_ws;                                         // [N, F_OUT]
    float* dinv = hw + (size_t)N_NODES * F_OUT;                         // [N]

    const int T = 256;
    const int gN      = (N_NODES + T - 1) / T;
    const int gE      = (N_EDGES + T - 1) / T;
    const int gNF     = (N_NODES * F_OUT + T - 1) / T;
    const long long ef = (long long)N_EDGES * F_OUT;
    const int gEF     = (int)((ef + T - 1) / T);
    const int gGEMM   = (M_TILES * 32 + T - 1) / T;                     // 8 waves/block

    for (int r = 0; r < N_REL; ++r) {
        const int*   src  = edges + (size_t)r * 2 * N_EDGES;
        const int*   dst  = src + N_EDGES;
        const int*   perm = perms + (size_t)r * N_NODES;
        const float* W    = Ws + (size_t)r * F_IN * F_OUT;
        const float* b    = bs + (size_t)r * F_OUT;
        float* pos  = pos_base + (size_t)r * N_NODES * F_OUT;
        float* neg  = neg_base + (size_t)r * N_NODES * F_OUT;
        float* summ = sum_base + (size_t)r * F_OUT;

        dmgi_deg_init <<<gN, T, 0, stream>>>(dinv);
        dmgi_deg_count<<<gE, T, 0, stream>>>(dst, dinv);
        dmgi_deg_rsqrt<<<gN, T, 0, stream>>>(dinv);

        dmgi_gemm_wmma<<<gGEMM, T, 0, stream>>>(x, W, hw);

        dmgi_init_out<<<gNF, T, 0, stream>>>(hw, dinv, perm, b, pos, neg, summ);
        dmgi_scatter <<<gEF, T, 0, stream>>>(hw, dinv, src, dst, perm, pos, neg);

        dmgi_relu<<<gNF, T, 0, stream>>>(neg, N_NODES * F_OUT);
        dmgi_relu_pos_summary<<<64, T, 0, stream>>>(pos, summ);
    }
}